// GCN_45140106281004
// MI455X (gfx1250) — compile-verified
//
#include <hip/hip_runtime.h>
#include <math.h>

typedef __attribute__((ext_vector_type(2))) float v2f;
typedef __attribute__((ext_vector_type(8))) float v8f;

#define N_NODES 100000
#define N_EDGES 1600000
#define NFEAT 128
#define NHID 64
#define NCLASS 40
#define NL 6

// ---------------------------------------------------------------------------
// Dense GEMM via V_WMMA_F32_16X16X4_F32, LDS-staged, compile-time shapes.
// OUT[N,KOUT] = H[N,K] @ W[K,KOUT]   (row-major f32, exact vs reference)
// block = 128 threads = 4 waves; wave w owns the 16x16 tile at
// rows [blockIdx.x*16,+16), cols [w*16,+16).  N % 16 == 0.
// W is staged into LDS zero-padded to 64 cols -> no masking in the K loop.
// ---------------------------------------------------------------------------
template <int K, int KOUT>
__global__ void gcn_gemm_wmma(const float* __restrict__ H,
                              const float* __restrict__ W,
                              float* __restrict__ OUT) {
  __shared__ float Bs[K * 64];   // [K][64], cols >= KOUT zero-filled
  __shared__ float As[16 * K];   // [16][K]

  const int tid  = threadIdx.x;          // 0..127
  const int row0 = blockIdx.x * 16;

  // stage W (zero-pad masked columns) -- coalesced
#pragma unroll 4
  for (int idx = tid; idx < K * 64; idx += 128) {
    const int col  = idx & 63;
    const int krow = idx >> 6;
    Bs[idx] = (col < KOUT) ? W[krow * KOUT + col] : 0.0f;
  }
  // stage A tile (16 x K) -- coalesced along K
#pragma unroll 4
  for (int idx = tid; idx < 16 * K; idx += 128) {
    const int rr = idx / K;              // K = 64/128 -> shift
    const int kk = idx & (K - 1);
    As[idx] = H[(size_t)(row0 + rr) * K + kk];
  }
  __syncthreads();

  const int wave = tid >> 5;
  const int lane = tid & 31;
  const int m    = lane & 15;            // A row / B-D col within tile
  const int half = lane >> 4;            // selects K pair (A/B) and M+8 (D)
  const int col  = (wave << 4) + m;

  const float* __restrict__ arow = As + m * K;
  const float* __restrict__ bcol = Bs + col;

  v8f acc = {};
#pragma unroll
  for (int k0 = 0; k0 < K; k0 += 4) {
    const int ka = k0 + 2 * half;
    v2f a, b;
    a.x = arow[ka];                      // A 16x4: vgpr j <- H[m][ka+j]
    a.y = arow[ka + 1];
    b.x = bcol[ka << 6];                 // B 4x16: vgpr j <- W[ka+j][col]
    b.y = bcol[(ka + 1) << 6];
    acc = __builtin_amdgcn_wmma_f32_16x16x4_f32(
        false, a, false, b, (short)0, acc, false, false);
  }

  if (col < KOUT) {
#pragma unroll
    for (int r = 0; r < 8; ++r)
      OUT[(size_t)(row0 + r + 8 * half) * KOUT + col] = acc[r];
  }
}

// ---------------------------------------------------------------------------
// Zero-fill (agg must start at 0 each layer; ws is poisoned by harness).
// ---------------------------------------------------------------------------
__global__ void gcn_zero_f32(float* __restrict__ p, int n) {
  int i = blockIdx.x * blockDim.x + threadIdx.x;
  if (i < n) p[i] = 0.0f;
}

// ---------------------------------------------------------------------------
// SpMM scatter: agg[row[e], f] += val[e] * S[col[e], f]
// One wave32 per edge. Edge index made explicitly wave-uniform via
// readfirstlane so row/col/val loads scalarize (SMEM). Lane covers features
// {lane, lane+32}: coalesced gather + global_atomic_add_f32, L2-resident.
// ---------------------------------------------------------------------------
__global__ void gcn_spmm_scatter(const int* __restrict__ erow,
                                 const int* __restrict__ ecol,
                                 const float* __restrict__ eval,
                                 const float* __restrict__ S,
                                 float* __restrict__ agg,
                                 int E, int F) {
  const int wlocal = __builtin_amdgcn_readfirstlane(threadIdx.x >> 5);
  const int wid    = blockIdx.x * (blockDim.x >> 5) + wlocal;   // edge id
  const int lane   = threadIdx.x & 31;
  if (wid >= E) return;

  const int   c = ecol[wid];
  const int   r = erow[wid];
  const float v = eval[wid];
  const float* __restrict__ src = S   + (size_t)c * F;
  float* __restrict__       dst = agg + (size_t)r * F;

  atomicAdd(&dst[lane], v * src[lane]);              // F >= 32 always
  const int f2 = lane + 32;
  if (f2 < F) atomicAdd(&dst[f2], v * src[f2]);
}

// ---------------------------------------------------------------------------
// out = relu(agg + b) (+ resid)      [fused bias/activation/residual]
// ---------------------------------------------------------------------------
__global__ void gcn_bias_relu_res(const float* __restrict__ agg,
                                  const float* __restrict__ b,
                                  const float* __restrict__ resid,
                                  float* __restrict__ out,
                                  int n, int F) {
  int i = blockIdx.x * blockDim.x + threadIdx.x;
  if (i >= n) return;
  int f = i & (F - 1);                 // F == 64 here
  float v = agg[i] + b[f];
  v = v > 0.0f ? v : 0.0f;
  if (resid) v += resid[i];
  out[i] = v;
}

// ---------------------------------------------------------------------------
// Final: out = log_softmax(agg + b2) over 40 classes, one thread per node.
// ---------------------------------------------------------------------------
__global__ void gcn_logsoftmax40(const float* __restrict__ agg,
                                 const float* __restrict__ b,
                                 float* __restrict__ out, int N) {
  int nidx = blockIdx.x * blockDim.x + threadIdx.x;
  if (nidx >= N) return;
  float vals[NCLASS];
  float mx = -INFINITY;
#pragma unroll
  for (int j = 0; j < NCLASS; ++j) {
    float v = agg[(size_t)nidx * NCLASS + j] + b[j];
    vals[j] = v;
    mx = fmaxf(mx, v);
  }
  float s = 0.0f;
#pragma unroll
  for (int j = 0; j < NCLASS; ++j) s += __expf(vals[j] - mx);
  float lse = mx + __logf(s);
#pragma unroll
  for (int j = 0; j < NCLASS; ++j)
    out[(size_t)nidx * NCLASS + j] = vals[j] - lse;
}

// ---------------------------------------------------------------------------
// Host-side per-layer sequence (all on `stream`, graph-capture safe)
// ---------------------------------------------------------------------------
static void run_sparse_part(float* support, float* agg,
                            const int* erow, const int* ecol, const float* eval,
                            int F, hipStream_t stream) {
  const int nagg = N_NODES * F;
  gcn_zero_f32<<<(nagg + 255) / 256, 256, 0, stream>>>(agg, nagg);
  // one wave per edge, 8 waves per 256-thread block
  gcn_spmm_scatter<<<(N_EDGES + 7) / 8, 256, 0, stream>>>(
      erow, ecol, eval, support, agg, N_EDGES, F);
}

extern "C" void kernel_launch(void* const* d_in, const int* in_sizes, int n_in,
                              void* d_out, int out_size, void* d_ws, size_t ws_size,
                              hipStream_t stream) {
  const float* x    = (const float*)d_in[0];
  const int*   erow = (const int*)d_in[1];
  const int*   ecol = (const int*)d_in[2];
  const float* eval = (const float*)d_in[3];
  const float* W1   = (const float*)d_in[4];
  const float* b1   = (const float*)d_in[5];
  const float* Wm   = (const float*)d_in[6];   // [6,64,64]
  const float* bm   = (const float*)d_in[7];   // [6,64]
  const float* W2   = (const float*)d_in[8];
  const float* b2   = (const float*)d_in[9];
  float* out = (float*)d_out;

  // workspace layout: 4 buffers of N*64 f32 (~102 MB total)
  float* support = (float*)d_ws;
  float* agg     = support + (size_t)N_NODES * NHID;
  float* bufA    = agg + (size_t)N_NODES * NHID;    // prev2 (h1)
  float* bufB    = bufA + (size_t)N_NODES * NHID;   // prev  (h2)

  const int nh = N_NODES * NHID;
  const int GEMM_GRID = N_NODES / 16;

  // ---- gc1: h1 = relu(spmm(x@W1) + b1) -> bufA
  gcn_gemm_wmma<NFEAT, NHID><<<GEMM_GRID, 128, 0, stream>>>(x, W1, support);
  run_sparse_part(support, agg, erow, ecol, eval, NHID, stream);
  gcn_bias_relu_res<<<(nh + 255) / 256, 256, 0, stream>>>(
      agg, b1, nullptr, bufA, nh, NHID);

  // ---- middle layer 0: h2 = relu(gconv(h1)) -> bufB (no residual)
  gcn_gemm_wmma<NHID, NHID><<<GEMM_GRID, 128, 0, stream>>>(bufA, Wm, support);
  run_sparse_part(support, agg, erow, ecol, eval, NHID, stream);
  gcn_bias_relu_res<<<(nh + 255) / 256, 256, 0, stream>>>(
      agg, bm, nullptr, bufB, nh, NHID);

  // ---- layers 1..5: h = relu(gconv(prev)) + prev2 ; rotate
  float* prev2 = bufA;
  float* prev  = bufB;
  for (int k = 1; k < NL; ++k) {
    gcn_gemm_wmma<NHID, NHID><<<GEMM_GRID, 128, 0, stream>>>(
        prev, Wm + (size_t)k * NHID * NHID, support);
    run_sparse_part(support, agg, erow, ecol, eval, NHID, stream);
    gcn_bias_relu_res<<<(nh + 255) / 256, 256, 0, stream>>>(
        agg, bm + (size_t)k * NHID, prev2, prev2, nh, NHID);
    float* t = prev2; prev2 = prev; prev = t;   // new prev = freshly written h
  }

  // ---- gc2: out = log_softmax(spmm(prev@W2) + b2)
  gcn_gemm_wmma<NHID, NCLASS><<<GEMM_GRID, 128, 0, stream>>>(prev, W2, support);
  run_sparse_part(support, agg, erow, ecol, eval, NCLASS, stream);
  gcn_logsoftmax40<<<(N_NODES + 127) / 128, 128, 0, stream>>>(agg, b2, out, N_NODES);
}